// GatedEnergySAGE_8667244003578
// MI455X (gfx1250) — compile-verified
//
#include <hip/hip_runtime.h>
#include <math.h>

// ---------------------------------------------------------------------------
// GatedEnergySAGE for MI455X (gfx1250, wave32).
// GEMMs via V_WMMA_F32_16X16X4_F32 (fp32 precision; workload is memory bound:
// ~23 GFLOP of GEMM vs ~2+GB irregular edge traffic at 23.3 TB/s HBM, so the
// fp32 matrix path is more than sufficient and keeps full accuracy).
// GEMM v2: A panels staged in LDS (shared by all waves of a block), branch-free
// clamped B column loads (no EXEC divergence in the inner loop).
// ---------------------------------------------------------------------------

typedef __attribute__((ext_vector_type(2))) float v2f;
typedef __attribute__((ext_vector_type(8))) float v8f;

#define FDIM 64
#define GHDIM 16
#define HDIM 256
#define CDIM 8
#define KMAX 256
#define LPAD 4  // LDS row padding (floats): stride K+4 keeps banks distinct

#define CDIV(a, b) (((a) + (b) - 1) / (b))

static __device__ __forceinline__ float sigmoidf_(float x) {
    return 1.0f / (1.0f + __expf(-x));
}

// ---------------- degree ----------------------------------------------------
__global__ void k_deg(const int* __restrict__ dst, float* __restrict__ deg, int E) {
    int i = blockIdx.x * blockDim.x + threadIdx.x;
    if (i < E) atomicAdd(&deg[dst[i]], 1.0f);
}

// xn = x * rsqrt(max(deg, 1e-12)) ------------------------------------------
__global__ void k_xn(const float* __restrict__ x, const float* __restrict__ deg,
                     float* __restrict__ xn, int total, int F) {
    int i = blockIdx.x * blockDim.x + threadIdx.x;
    if (i < total) {
        int r = i / F;
        xn[i] = x[i] * rsqrtf(fmaxf(deg[r], 1e-12f));
    }
}

// per-edge energy accumulation (4 features per thread, float4 gathers) ------
__global__ void k_energy(const int* __restrict__ src, const int* __restrict__ dst,
                         const float* __restrict__ xn,
                         float* __restrict__ num, float* __restrict__ den, int E) {
    int i = blockIdx.x * blockDim.x + threadIdx.x;
    int total = E * (FDIM / 4);
    if (i >= total) return;
    int e = i / (FDIM / 4);
    int f = (i % (FDIM / 4)) * 4;
    int s = src[e], d = dst[e];
    const float4 xs = *(const float4*)(xn + (size_t)s * FDIM + f);
    const float4 xd = *(const float4*)(xn + (size_t)d * FDIM + f);
    float dx0 = xd.x - xs.x, dx1 = xd.y - xs.y, dx2 = xd.z - xs.z, dx3 = xd.w - xs.w;
    size_t base = (size_t)d * FDIM + f;
    atomicAdd(&num[base + 0], dx0 * dx0);
    atomicAdd(&num[base + 1], dx1 * dx1);
    atomicAdd(&num[base + 2], dx2 * dx2);
    atomicAdd(&num[base + 3], dx3 * dx3);
    atomicAdd(&den[base + 0], xd.x * xd.x + xs.x * xs.x);
    atomicAdd(&den[base + 1], xd.y * xd.y + xs.y * xs.y);
    atomicAdd(&den[base + 2], xd.z * xd.z + xs.z * xs.z);
    atomicAdd(&den[base + 3], xd.w * xd.w + xs.w * xs.w);
}

// R = num / (den + 1e-8), in place in num -----------------------------------
__global__ void k_finalize_R(float* __restrict__ num, const float* __restrict__ den,
                             int total) {
    int i = blockIdx.x * blockDim.x + threadIdx.x;
    if (i < total) num[i] = num[i] / (den[i] + 1e-8f);
}

// column mean/std (ddof=1); one block per column ----------------------------
__global__ void k_colstats(const float* __restrict__ x, float* __restrict__ mean,
                           float* __restrict__ stdv, int M, int F) {
    int f = blockIdx.x;
    float s = 0.0f, s2 = 0.0f;
    for (int r = threadIdx.x; r < M; r += blockDim.x) {
        float v = x[(size_t)r * F + f];
        s += v;
        s2 += v * v;
    }
    __shared__ float sh[256];
    __shared__ float sh2[256];
    sh[threadIdx.x] = s;
    sh2[threadIdx.x] = s2;
    __syncthreads();
    for (int w = blockDim.x >> 1; w > 0; w >>= 1) {
        if ((int)threadIdx.x < w) {
            sh[threadIdx.x] += sh[threadIdx.x + w];
            sh2[threadIdx.x] += sh2[threadIdx.x + w];
        }
        __syncthreads();
    }
    if (threadIdx.x == 0) {
        float m = sh[0] / (float)M;
        float var = (sh2[0] - sh[0] * sh[0] / (float)M) / (float)(M - 1);
        mean[f] = m;
        stdv[f] = fmaxf(sqrtf(fmaxf(var, 0.0f)), 1e-8f);
    }
}

// out = (x - mean[col]) / std[col] (in-place safe) --------------------------
__global__ void k_znorm(const float* __restrict__ x, const float* __restrict__ mean,
                        const float* __restrict__ stdv, float* __restrict__ out,
                        int total, int F) {
    int i = blockIdx.x * blockDim.x + threadIdx.x;
    if (i < total) {
        int f = i % F;
        out[i] = (x[i] - mean[f]) / stdv[f];
    }
}

// Z = sig(g2)*Rn + (1-sig(g2))*Rf with Rn=(R-rm)/rs, Rf=(2-R-rm)/rs ---------
__global__ void k_gateZ(const float* __restrict__ g2, const float* __restrict__ R,
                        const float* __restrict__ rm, const float* __restrict__ rs,
                        float* __restrict__ Z, int total, int F) {
    int i = blockIdx.x * blockDim.x + threadIdx.x;
    if (i < total) {
        int f = i % F;
        float g = sigmoidf_(g2[i]);
        float r = R[i];
        float rn = (r - rm[f]) / rs[f];
        float rf = ((2.0f - r) - rm[f]) / rs[f];
        Z[i] = g * rn + (1.0f - g) * rf;
    }
}

// h = en * sigmoid(a2) -------------------------------------------------------
__global__ void k_mulsig(const float* __restrict__ en, const float* __restrict__ a2,
                         float* __restrict__ h, int total) {
    int i = blockIdx.x * blockDim.x + threadIdx.x;
    if (i < total) h[i] = en[i] * sigmoidf_(a2[i]);
}

// scatter-add neighbor features: agg[dst] += h[src] (4 feats/thread) --------
__global__ void k_aggregate(const int* __restrict__ src, const int* __restrict__ dst,
                            const float* __restrict__ h, float* __restrict__ agg,
                            int E, int dim) {
    int i = blockIdx.x * blockDim.x + threadIdx.x;
    int chunks = dim >> 2;
    int total = E * chunks;
    if (i >= total) return;
    int e = i / chunks;
    int f = (i - e * chunks) * 4;
    int s = src[e], d = dst[e];
    const float4 v = *(const float4*)(h + (size_t)s * dim + f);
    size_t base = (size_t)d * dim + f;
    atomicAdd(&agg[base + 0], v.x);
    atomicAdd(&agg[base + 1], v.y);
    atomicAdd(&agg[base + 2], v.z);
    atomicAdd(&agg[base + 3], v.w);
}

// agg /= max(deg, 1) ---------------------------------------------------------
__global__ void k_scaledeg(float* __restrict__ agg, const float* __restrict__ deg,
                           int total, int dim) {
    int i = blockIdx.x * blockDim.x + threadIdx.x;
    if (i < total) {
        int r = i / dim;
        agg[i] = agg[i] / fmaxf(deg[r], 1.0f);
    }
}

// ---------------------------------------------------------------------------
// WMMA GEMM v2: out[M,Nout] = (relu?)( A1@W1 + (A2?A2@W2:0) + bias ), fp32.
// Grid: blockIdx.y = M-tile (16 rows), waves of blockIdx.x = N-tiles.
// A panels (16 x K) staged cooperatively into LDS (row stride K+LPAD so LDS
// banks (stride*m mod 64 = 4m) stay distinct across the 16 rows).
// B columns for lanes past Nout are CLAMPED, not masked: column n of C depends
// only on column n of B, and out-of-range columns are never stored, so the
// inner loop is completely branch-free (no EXEC manipulation around v_wmma).
// Fragment layouts per ISA 7.12.2 (f32):
//   A 16x4 : lane m = lane&15, k-half = 2*(lane>>4), regs {k, k+1}
//   B 4x16 : lane n = lane&15, same k-half split across lane halves
//   C 16x16: reg r = row r (lanes 0-15) / row r+8 (lanes 16-31)
// ---------------------------------------------------------------------------
__global__ void k_gemm_wmma(const float* __restrict__ A1, const float* __restrict__ W1,
                            const float* __restrict__ A2, const float* __restrict__ W2,
                            const float* __restrict__ bias, float* __restrict__ out,
                            int M, int K, int Nout, int doRelu) {
    __shared__ float lA[2 * 16 * (KMAX + LPAD)];
    float* lA1 = lA;
    float* lA2 = lA + 16 * (KMAX + LPAD);
    const int lstr = K + LPAD;
    const int tm = blockIdx.y;

    // cooperative float4 staging of the 16 x K A panels
    const int nchunks = 4 * K;  // 16 rows * (K/4) float4s
    const int kc = K >> 2;
    for (int idx = threadIdx.x; idx < nchunks; idx += blockDim.x) {
        int rrow = idx / kc;
        int rcol = (idx - rrow * kc) << 2;
        float4 v = *(const float4*)(A1 + ((size_t)(tm * 16 + rrow)) * K + rcol);
        *(float4*)(lA1 + rrow * lstr + rcol) = v;
    }
    if (A2 != nullptr) {
        for (int idx = threadIdx.x; idx < nchunks; idx += blockDim.x) {
            int rrow = idx / kc;
            int rcol = (idx - rrow * kc) << 2;
            float4 v = *(const float4*)(A2 + ((size_t)(tm * 16 + rrow)) * K + rcol);
            *(float4*)(lA2 + rrow * lstr + rcol) = v;
        }
    }
    __syncthreads();

    const int wave = threadIdx.x >> 5;
    const int lane = threadIdx.x & 31;
    const int ntn = (Nout + 15) >> 4;
    const int tn = blockIdx.x * (blockDim.x >> 5) + wave;
    if (tn >= ntn) return;  // wave-uniform; no further syncthreads

    const int col = (tn << 4) + (lane & 15);
    const int colc = (col < Nout) ? col : (Nout - 1);  // clamped, branch-free
    const int kh = (lane >> 4) << 1;                   // 0 or 2
    const float* la1 = lA1 + (lane & 15) * lstr + kh;
    const float* la2 = lA2 + (lane & 15) * lstr + kh;
    const float* w1p = W1 + colc;
    v8f acc = {};

    #pragma unroll 4
    for (int k0 = 0; k0 < K; k0 += 4) {
        v2f a = {la1[k0], la1[k0 + 1]};
        v2f b = {w1p[(size_t)(k0 + kh) * Nout], w1p[(size_t)(k0 + kh + 1) * Nout]};
        acc = __builtin_amdgcn_wmma_f32_16x16x4_f32(false, a, false, b, (short)0, acc,
                                                    false, false);
    }
    if (A2 != nullptr) {
        const float* w2p = W2 + colc;
        #pragma unroll 4
        for (int k0 = 0; k0 < K; k0 += 4) {
            v2f a = {la2[k0], la2[k0 + 1]};
            v2f b = {w2p[(size_t)(k0 + kh) * Nout], w2p[(size_t)(k0 + kh + 1) * Nout]};
            acc = __builtin_amdgcn_wmma_f32_16x16x4_f32(false, a, false, b, (short)0,
                                                        acc, false, false);
        }
    }

    if (col < Nout) {
        float bv = bias[col];
        int rbase = (tm << 4) + ((lane >> 4) << 3);
        #pragma unroll
        for (int r = 0; r < 8; ++r) {
            float v = acc[r] + bv;
            if (doRelu) v = fmaxf(v, 0.0f);
            out[(size_t)(rbase + r) * Nout + col] = v;
        }
    }
}

// ---------------------------------------------------------------------------
extern "C" void kernel_launch(void* const* d_in, const int* in_sizes, int n_in,
                              void* d_out, int out_size, void* d_ws, size_t ws_size,
                              hipStream_t stream) {
    const float* features = (const float*)d_in[0];
    const int* edge_index = (const int*)d_in[1];
    const float* gate_w1 = (const float*)d_in[2];
    const float* gate_b1 = (const float*)d_in[3];
    const float* gate_w2 = (const float*)d_in[4];
    const float* gate_b2 = (const float*)d_in[5];
    const float* attn_w1 = (const float*)d_in[6];
    const float* attn_b1 = (const float*)d_in[7];
    const float* attn_w2 = (const float*)d_in[8];
    const float* attn_b2 = (const float*)d_in[9];
    const float* c1_ws = (const float*)d_in[10];
    const float* c1_wn = (const float*)d_in[11];
    const float* c1_b = (const float*)d_in[12];
    const float* c2_ws = (const float*)d_in[13];
    const float* c2_wn = (const float*)d_in[14];
    const float* c2_b = (const float*)d_in[15];
    const float* c3_ws = (const float*)d_in[16];
    const float* c3_wn = (const float*)d_in[17];
    const float* c3_b = (const float*)d_in[18];
    const float* cls_w = (const float*)d_in[19];
    const float* cls_b = (const float*)d_in[20];

    const int N = in_sizes[0] / FDIM;  // 50000 (multiple of 16)
    const int E = in_sizes[1] / 2;     // 800000
    const int* src = edge_index;
    const int* dst = edge_index + E;

    // ----- workspace layout (floats) -----
    float* W = (float*)d_ws;
    size_t o = 0;
    float* stats = W; o += 512;  // fmean/fstd/rm/rs/zm/zs (64 each)
    float* fmean = stats + 0;
    float* fstd = stats + 64;
    float* rm = stats + 128;
    float* rs = stats + 192;
    float* zm = stats + 256;
    float* zs = stats + 320;
    float* deg = W + o; o += (size_t)N;
    float* xn = W + o; o += (size_t)N * FDIM;   // xn, then reused as Xn
    float* num = W + o; o += (size_t)N * FDIM;  // energy num, then R
    float* den = W + o; o += (size_t)N * FDIM;  // energy den, then g2/a2 scratch
    float* g1 = W + o; o += (size_t)N * GHDIM;  // gate/attn hidden
    float* Zb = W + o; o += (size_t)N * FDIM;   // Z, then en (in-place norm)
    float* h0 = W + o; o += (size_t)N * FDIM;   // attention-weighted input
    float* agg = W + o; o += (size_t)N * HDIM;  // neighbor mean buffer
    float* h1 = W + o; o += (size_t)N * HDIM;   // layer1 out; reused for layer3 out
    float* h2 = W + o; o += (size_t)N * HDIM;   // layer2 out
    float* outp = (float*)d_out;

    const int B = 256;

    auto launch_gemm = [&](const float* A1, const float* W1, const float* A2,
                           const float* W2, const float* bias, float* out, int M, int K,
                           int Nout, int relu) {
        int ntn = CDIV(Nout, 16);
        int wpb = ntn < 8 ? ntn : 8;  // waves per block
        dim3 grid(CDIV(ntn, wpb), M / 16);
        k_gemm_wmma<<<grid, wpb * 32, 0, stream>>>(A1, W1, A2, W2, bias, out, M, K,
                                                   Nout, relu);
    };

    // ----- zero accumulation buffers (graph-capturable memsets) -----
    hipMemsetAsync(deg, 0, sizeof(float) * (size_t)N, stream);
    hipMemsetAsync(num, 0, sizeof(float) * (size_t)N * FDIM, stream);
    hipMemsetAsync(den, 0, sizeof(float) * (size_t)N * FDIM, stream);

    // ----- degree + local energy -----
    k_deg<<<CDIV(E, B), B, 0, stream>>>(dst, deg, E);
    k_xn<<<CDIV(N * FDIM, B), B, 0, stream>>>(features, deg, xn, N * FDIM, FDIM);
    k_energy<<<CDIV(E * (FDIM / 4), B), B, 0, stream>>>(src, dst, xn, num, den, E);
    k_finalize_R<<<CDIV(N * FDIM, B), B, 0, stream>>>(num, den, N * FDIM);

    // ----- z-stats -----
    k_colstats<<<FDIM, B, 0, stream>>>(features, fmean, fstd, N, FDIM);
    k_colstats<<<FDIM, B, 0, stream>>>(num, rm, rs, N, FDIM);
    k_znorm<<<CDIV(N * FDIM, B), B, 0, stream>>>(features, fmean, fstd, xn, N * FDIM,
                                                 FDIM);

    // ----- gate MLP: g1 = relu(Xn@W1+b1); g2 = g1@W2+b2; Z = gate-mix -----
    launch_gemm(xn, gate_w1, nullptr, nullptr, gate_b1, g1, N, FDIM, GHDIM, 1);
    launch_gemm(g1, gate_w2, nullptr, nullptr, gate_b2, den, N, GHDIM, FDIM, 0);
    k_gateZ<<<CDIV(N * FDIM, B), B, 0, stream>>>(den, num, rm, rs, Zb, N * FDIM, FDIM);

    // ----- en = zscore(Z); attn MLP; h0 = en * sigmoid(attn) -----
    k_colstats<<<FDIM, B, 0, stream>>>(Zb, zm, zs, N, FDIM);
    k_znorm<<<CDIV(N * FDIM, B), B, 0, stream>>>(Zb, zm, zs, Zb, N * FDIM, FDIM);
    launch_gemm(Zb, attn_w1, nullptr, nullptr, attn_b1, g1, N, FDIM, GHDIM, 1);
    launch_gemm(g1, attn_w2, nullptr, nullptr, attn_b2, den, N, GHDIM, FDIM, 0);
    k_mulsig<<<CDIV(N * FDIM, B), B, 0, stream>>>(Zb, den, h0, N * FDIM);

    // ----- SAGE layer 1: 64 -> 256 -----
    hipMemsetAsync(agg, 0, sizeof(float) * (size_t)N * FDIM, stream);
    k_aggregate<<<CDIV(E * (FDIM / 4), B), B, 0, stream>>>(src, dst, h0, agg, E, FDIM);
    k_scaledeg<<<CDIV(N * FDIM, B), B, 0, stream>>>(agg, deg, N * FDIM, FDIM);
    launch_gemm(h0, c1_ws, agg, c1_wn, c1_b, h1, N, FDIM, HDIM, 1);

    // ----- SAGE layer 2: 256 -> 256 -----
    hipMemsetAsync(agg, 0, sizeof(float) * (size_t)N * HDIM, stream);
    k_aggregate<<<CDIV(E * (HDIM / 4), B), B, 0, stream>>>(src, dst, h1, agg, E, HDIM);
    k_scaledeg<<<CDIV(N * HDIM, B), B, 0, stream>>>(agg, deg, N * HDIM, HDIM);
    launch_gemm(h1, c2_ws, agg, c2_wn, c2_b, h2, N, HDIM, HDIM, 1);

    // ----- SAGE layer 3: 256 -> 128 (output into h1 buffer) -----
    hipMemsetAsync(agg, 0, sizeof(float) * (size_t)N * HDIM, stream);
    k_aggregate<<<CDIV(E * (HDIM / 4), B), B, 0, stream>>>(src, dst, h2, agg, E, HDIM);
    k_scaledeg<<<CDIV(N * HDIM, B), B, 0, stream>>>(agg, deg, N * HDIM, HDIM);
    launch_gemm(h2, c3_ws, agg, c3_wn, c3_b, h1, N, HDIM, HDIM / 2, 1);

    // ----- classifier: 128 -> 8 -----
    launch_gemm(h1, cls_w, nullptr, nullptr, cls_b, outp, N, HDIM / 2, CDIM, 0);
}